// WeightedLoss_53858889891883
// MI455X (gfx1250) — compile-verified
//
#include <hip/hip_runtime.h>
#include <math.h>

// ---------------------------------------------------------------------------
// Adaptive Wing loss with cv2-style 3x3 dilation mask, mean-reduced.
// Shapes: pred/landmark (64,68,128,128) f32.  ~570MB HBM reads -> ~24.5us
// roofline at 23.3 TB/s.  One block per 128x128 image; landmark staged in
// LDS via CDNA5 GLOBAL_LOAD_ASYNC_TO_LDS_B128.  Each wave-iteration covers
// one full image row (32 lanes x float4), so the 3x3 dilation needs only
// 3 ds_load_b128 + 2 cross-lane shuffles per quad.  All transcendentals use
// raw v_log_f32/v_exp_f32/v_rcp_f32 to keep VALU work ~20 ops/px.
// ---------------------------------------------------------------------------

#define IMG_H 128
#define IMG_W 128
#define IMG_PX (IMG_H * IMG_W)        // 16384 px, 64KB f32
#define BLOCK 256
#define QUADS_PER_IMG (IMG_PX / 4)    // 4096 float4 quads
#define QPT (QUADS_PER_IMG / BLOCK)   // 16 quads per thread

// --- CDNA5 async global->LDS path (guarded; falls back to ds stores) -------
#if defined(__HIP_DEVICE_COMPILE__) && defined(__has_builtin)
#if __has_builtin(__builtin_amdgcn_global_load_async_to_lds_b128) && \
    __has_builtin(__builtin_amdgcn_s_wait_asynccnt)
#define USE_ASYNC_LDS 1
#endif
#if __has_builtin(__builtin_amdgcn_logf) && \
    __has_builtin(__builtin_amdgcn_exp2f) && \
    __has_builtin(__builtin_amdgcn_rcpf)
#define USE_HW_TRANS 1
#endif
#endif
#ifndef USE_ASYNC_LDS
#define USE_ASYNC_LDS 0
#endif
#ifndef USE_HW_TRANS
#define USE_HW_TRANS 0
#endif

#if USE_ASYNC_LDS
// Builtin signature (from clang diagnostic): pointers to 16-byte int vectors,
// global (AS1) source, LDS (AS3) destination, two int immediates.
typedef int v4i_t __attribute__((vector_size(4 * sizeof(int))));
typedef __attribute__((address_space(1))) v4i_t* gptr_v4i;
typedef __attribute__((address_space(3))) v4i_t* lptr_v4i;
#endif

__device__ __forceinline__ float fast_log2(float x) {
#if USE_HW_TRANS
  return __builtin_amdgcn_logf(x);     // v_log_f32: log2(0) = -inf
#else
  return log2f(x);
#endif
}
__device__ __forceinline__ float fast_exp2(float x) {
#if USE_HW_TRANS
  return __builtin_amdgcn_exp2f(x);    // v_exp_f32: exp2(-inf) = 0
#else
  return exp2f(x);
#endif
}
__device__ __forceinline__ float fast_rcp(float x) {
#if USE_HW_TRANS
  return __builtin_amdgcn_rcpf(x);     // v_rcp_f32, ~1 ulp
#else
  return 1.0f / x;
#endif
}

// ALPHA=2.1, OMEGA=14, EPSILON=1, THETA=0.5
__device__ __forceinline__ float awing_loss(float p, float y) {
  const float LN2 = 0.69314718055994531f;
  const float d    = fabsf(p - y);
  const float asl  = 2.1f - y;                          // in [1.1, 2.1]
  const float T1   = fast_exp2(-asl);                   // 0.5^asl in [0.23,0.47]
  const float A    = 28.0f * asl * T1 * fast_rcp(1.0f + T1);
  // C = 0.5*A - 14*log1p(T1); 1+T1 >= 1.23 so plain log is safe.
  const float C    = fmaf(-(14.0f * LN2), fast_log2(1.0f + T1), 0.5f * A);
  // d^asl = exp2(asl*log2(d)); d==0 -> -inf -> 0. Result in [0, 0.47).
  const float powd = fast_exp2(asl * fast_log2(d));
  const float lsml = (14.0f * LN2) * fast_log2(1.0f + powd);
  const float llin = fmaf(A, d, -C);
  return d < 0.5f ? lsml : llin;
}

// mask weight: round(dil*255) >= 128*0.2 (=25.6)  ->  W_MASK*M+1 = 11 or 1
__device__ __forceinline__ float mask_w(float dilraw) {
  return (rintf(dilraw * 255.0f) >= 25.6f) ? 11.0f : 1.0f;
}

template <bool ATOMIC>
__global__ void __launch_bounds__(BLOCK)
awing_kernel(const float* __restrict__ pred, const float* __restrict__ lm,
             float* __restrict__ partial, float* __restrict__ outAtomic,
             float scale) {
  __shared__ __align__(16) float s_lm[IMG_PX];   // 64KB: one landmark image

  const int img = blockIdx.x;
  const int t   = threadIdx.x;
  const int L   = t & 31;                        // lane within wave
  const int wv  = t >> 5;                        // wave id (0..7)
  const float* __restrict__ lmImg = lm   + (size_t)img * IMG_PX;
  const float* __restrict__ prImg = pred + (size_t)img * IMG_PX;

  // ---- Stage landmark image into LDS (16B per lane per issue) ----
#if USE_ASYNC_LDS
  #pragma unroll
  for (int j = 0; j < QPT; ++j) {
    const int q = t + j * BLOCK;                 // float4 index 0..4095
    __builtin_amdgcn_global_load_async_to_lds_b128(
        (gptr_v4i)(lmImg + 4 * q),
        (lptr_v4i)(&s_lm[4 * q]),
        0, 0);
  }
  __builtin_amdgcn_s_wait_asynccnt(0);
#else
  #pragma unroll
  for (int j = 0; j < QPT; ++j) {
    const int q = t + j * BLOCK;
    ((float4*)s_lm)[q] = ((const float4*)lmImg)[q];
  }
#endif
  __syncthreads();

  const float NEG = -3.0e38f;                    // -inf padding (SAME)
  const float4* __restrict__ s4 = (const float4*)s_lm;

  float acc = 0.0f;
  // Each wave-iteration covers one full image row: lane L -> cols 4L..4L+3,
  // row r = wv + 8*j (wave-uniform -> scalar row-bound branches).
  #pragma unroll
  for (int j = 0; j < QPT; ++j) {
    const int r = wv + 8 * j;                    // 0..127
    const int q = (r << 5) + L;                  // float4 index == t + j*256

    const float4 p4 = ((const float4*)prImg)[q]; // coalesced b128 load
    const float4 c4 = s4[q];                     // center row (landmark vals)
    float4 u4, d4;
    if (r > 0)         u4 = s4[q - 32]; else u4 = make_float4(NEG, NEG, NEG, NEG);
    if (r < IMG_H - 1) d4 = s4[q + 32]; else d4 = make_float4(NEG, NEG, NEG, NEG);

    // vertical 3-max per column (v_max3)
    const float vm0 = fmaxf(fmaxf(c4.x, u4.x), d4.x);
    const float vm1 = fmaxf(fmaxf(c4.y, u4.y), d4.y);
    const float vm2 = fmaxf(fmaxf(c4.z, u4.z), d4.z);
    const float vm3 = fmaxf(fmaxf(c4.w, u4.w), d4.w);

    // boundary columns from adjacent lanes; wave spans the whole row, so
    // lane edges coincide with image edges.
    float vmL = __shfl_up(vm3, 1);
    float vmR = __shfl_down(vm0, 1);
    if (L == 0)  vmL = NEG;
    if (L == 31) vmR = NEG;

    const float dil0 = fmaxf(fmaxf(vmL, vm0), vm1);
    const float dil1 = fmaxf(fmaxf(vm0, vm1), vm2);
    const float dil2 = fmaxf(fmaxf(vm1, vm2), vm3);
    const float dil3 = fmaxf(fmaxf(vm2, vm3), vmR);

    acc = fmaf(awing_loss(p4.x, c4.x), mask_w(dil0), acc);
    acc = fmaf(awing_loss(p4.y, c4.y), mask_w(dil1), acc);
    acc = fmaf(awing_loss(p4.z, c4.z), mask_w(dil2), acc);
    acc = fmaf(awing_loss(p4.w, c4.w), mask_w(dil3), acc);
  }

  // ---- Block reduction (reuse s_lm after barrier) ----
  __syncthreads();
  s_lm[t] = acc;
  __syncthreads();
  #pragma unroll
  for (int s = BLOCK / 2; s > 0; s >>= 1) {
    if (t < s) s_lm[t] += s_lm[t + s];
    __syncthreads();
  }
  if (t == 0) {
    if (ATOMIC) {
      atomicAdd(outAtomic, s_lm[0] * scale);
    } else {
      partial[blockIdx.x] = s_lm[0];
    }
  }
}

// Final deterministic reduction of per-image partial sums (f64 accumulate).
__global__ void __launch_bounds__(BLOCK)
reduce_kernel(const float* __restrict__ partial, float* __restrict__ out,
              int n, double invN) {
  __shared__ double sd[BLOCK];
  const int t = threadIdx.x;
  double a = 0.0;
  for (int i = t; i < n; i += BLOCK) a += (double)partial[i];
  sd[t] = a;
  __syncthreads();
  #pragma unroll
  for (int s = BLOCK / 2; s > 0; s >>= 1) {
    if (t < s) sd[t] += sd[t + s];
    __syncthreads();
  }
  if (t == 0) out[0] = (float)(sd[0] * invN);
}

extern "C" void kernel_launch(void* const* d_in, const int* in_sizes, int n_in,
                              void* d_out, int out_size, void* d_ws, size_t ws_size,
                              hipStream_t stream) {
  const float* pred = (const float*)d_in[0];
  const float* lm   = (const float*)d_in[1];
  float* out        = (float*)d_out;

  const int    nTotal = in_sizes[0];             // 71,303,168
  const int    nImg   = nTotal / IMG_PX;         // 4352
  const double invN   = 1.0 / (double)nTotal;

  if (ws_size >= (size_t)nImg * sizeof(float)) {
    // Deterministic two-pass path.
    float* partial = (float*)d_ws;
    awing_kernel<false><<<nImg, BLOCK, 0, stream>>>(pred, lm, partial, nullptr, 0.0f);
    reduce_kernel<<<1, BLOCK, 0, stream>>>(partial, out, nImg, invN);
  } else {
    // Fallback: atomic accumulation directly into d_out.
    (void)hipMemsetAsync(out, 0, sizeof(float), stream);
    awing_kernel<true><<<nImg, BLOCK, 0, stream>>>(pred, lm, nullptr, out, (float)invN);
  }
}